// AffinityNet_59622736003105
// MI455X (gfx1250) — compile-verified
//
#include <hip/hip_runtime.h>

typedef __attribute__((ext_vector_type(2))) float v2f;
typedef __attribute__((ext_vector_type(8))) float v8f;

#define FDIM 128
#define G_NUM 128
#define BN_EPS 1e-5f
#define POOL_CHUNK 64
#define EDGES_PER_WAVE 4

// ---------------------------------------------------------------------------
// Utility: zero-fill (workspace is poisoned with 0xAA by the harness)
// ---------------------------------------------------------------------------
__global__ void fill_zero(float* __restrict__ p, int n) {
  int t = blockIdx.x * blockDim.x + threadIdx.x;
  int stride = gridDim.x * blockDim.x;
  for (int i = t; i < n; i += stride) p[i] = 0.0f;
}

// ---------------------------------------------------------------------------
// Degree: deg[dst] += 1 (self-loop +1 folded into dinv kernel)
// ---------------------------------------------------------------------------
__global__ void degree_kernel(const int* __restrict__ dst, float* __restrict__ deg, int E) {
  int e = blockIdx.x * blockDim.x + threadIdx.x;
  if (e < E) atomicAdd(&deg[dst[e]], 1.0f);
}

__global__ void dinv_kernel(float* __restrict__ deg_dinv, int N) {
  int i = blockIdx.x * blockDim.x + threadIdx.x;
  if (i < N) deg_dinv[i] = rsqrtf(deg_dinv[i] + 1.0f);   // (deg+self)^(-1/2)
}

// ---------------------------------------------------------------------------
// Y[N,128] = X[N,128] @ W[128,128] using V_WMMA_F32_16X16X4_F32 (exact fp32).
// One wave -> 16-row strip, 8 column tiles (8 x v8f accumulators).
//
// W is staged in LDS row-pair interleaved:
//   sW[p*256 + 2c + 0] = W[2p  ][c]
//   sW[p*256 + 2c + 1] = W[2p+1][c]
// so each B fragment (K-pair for one column) is ONE aligned ds_load_b64 that
// lands directly in an even VGPR pair for WMMA SRC1 (no register shuffles).
//
// A frag (16x4 f32): lane<16 -> (X[m][k0],X[m][k0+1]); lane>=16 -> K+2 pair.
// Out-of-range rows are CLAMPED on load (WMMA row M of D depends only on
// row M of A) and masked at store; k-loop stays branch-free, EXEC all-ones.
// Safe for in-place X==Y (each wave reads only its own rows, stores last).
// ---------------------------------------------------------------------------
__global__ void __launch_bounds__(256) gemm_wmma_f32(const float* __restrict__ X,
                                                     const float* __restrict__ W,
                                                     float* __restrict__ Y, int N) {
  __shared__ float sW[FDIM * FDIM];                       // 64 KB of 320 KB WGP LDS
  const int tid = threadIdx.x;

  // Stage W with row-pair interleave: task = (pair p, column-quad cq).
  // Reads are coalesced (two contiguous 16B segments per task); writes are
  // two contiguous float4s at sW + p*256 + cq*8.
#pragma unroll
  for (int it = 0; it < 8; ++it) {
    int task = tid + it * 256;                            // 0..2047
    int p  = task >> 5;                                   // K-pair 0..63
    int cq = task & 31;                                   // column quad 0..31
    float4 r0 = *(const float4*)(W + (size_t)(2 * p) * FDIM + cq * 4);
    float4 r1 = *(const float4*)(W + (size_t)(2 * p + 1) * FDIM + cq * 4);
    float4 w0 = {r0.x, r1.x, r0.y, r1.y};
    float4 w1 = {r0.z, r1.z, r0.w, r1.w};
    float4* dstp = (float4*)(sW + p * 256 + cq * 8);
    dstp[0] = w0;
    dstp[1] = w1;
  }
  __syncthreads();

  const int wave  = tid >> 5;
  const int lane  = tid & 31;
  const int mlane = lane & 15;
  const int koff  = (lane < 16) ? 0 : 2;                  // K-pair select per lane half
  const int m0    = blockIdx.x * 128 + wave * 16;
  const int rowA  = m0 + mlane;
  const int rowSafe = (rowA < N) ? rowA : (N - 1);        // clamp, mask at store
  const float* arow = X + (size_t)rowSafe * FDIM + koff;

  // Per-lane B base: pair offset (0 or 1) * 256 + column*2
  const float* brow = sW + ((lane < 16) ? 0 : 256) + mlane * 2;

  v8f acc[8] = {};                                        // 8 tiles of 16x16 f32

  for (int k0 = 0; k0 < FDIM; k0 += 4) {
    v2f a = *(const v2f*)(arow + k0);                     // unconditional 8B load
    const float* bk = brow + (k0 >> 1) * 256;             // advance 2 K-pairs
#pragma unroll
    for (int n = 0; n < 8; ++n) {
      v2f b = *(const v2f*)(bk + n * 32);                 // one ds_load_b64
      acc[n] = __builtin_amdgcn_wmma_f32_16x16x4_f32(
          /*neg_a=*/false, a, /*neg_b=*/false, b,
          /*c_mod=*/(short)0, acc[n], /*reuse_a=*/false, /*reuse_b=*/false);
    }
  }

  // C/D layout: VGPR r holds row M=r (lanes 0-15) / M=r+8 (lanes 16-31)
  const int rbase = m0 + ((lane < 16) ? 0 : 8);
#pragma unroll
  for (int n = 0; n < 8; ++n) {
    const int col = n * 16 + mlane;
#pragma unroll
    for (int r = 0; r < 8; ++r) {
      const int row = rbase + r;
      if (row < N) Y[(size_t)row * FDIM + col] = acc[n][r];
    }
  }
}

// ---------------------------------------------------------------------------
// Edge aggregation: agg[dst] += h[src] * dinv[src]*dinv[dst]
// One wave per 4 edges; each lane owns 4 features (float4 gather + 4 atomics).
// global_prefetch_b8 of the next edge's h[src] row hides gather latency
// behind the current edge's atomic traffic.
// ---------------------------------------------------------------------------
__global__ void scatter_kernel(const int* __restrict__ src, const int* __restrict__ dst,
                               const float* __restrict__ dinv, const float* __restrict__ h,
                               float* __restrict__ agg, int E) {
  int t = blockIdx.x * blockDim.x + threadIdx.x;
  int w = t >> 5, lane = t & 31;
  int e0 = w * EDGES_PER_WAVE;
  if (e0 >= E) return;
  int e1 = min(e0 + EDGES_PER_WAVE, E);

  __builtin_prefetch(h + (size_t)src[e0] * FDIM + lane * 4, 0, 0);

  for (int e = e0; e < e1; ++e) {
    if (e + 1 < e1)
      __builtin_prefetch(h + (size_t)src[e + 1] * FDIM + lane * 4, 0, 0);
    int s = src[e], d = dst[e];
    float wgt = dinv[s] * dinv[d];
    float4 v = ((const float4*)(h + (size_t)s * FDIM))[lane];
    float* a = agg + (size_t)d * FDIM + lane * 4;
    atomicAdd(a + 0, v.x * wgt);
    atomicAdd(a + 1, v.y * wgt);
    atomicAdd(a + 2, v.z * wgt);
    atomicAdd(a + 3, v.w * wgt);
  }
}

// h = relu(agg + h * dinv^2 + bias)  (in place on h)
__global__ void combine_kernel(float* __restrict__ h, const float* __restrict__ agg,
                               const float* __restrict__ dinv, const float* __restrict__ bias,
                               int N) {
  int t = blockIdx.x * blockDim.x + threadIdx.x;
  if (t >= N * FDIM) return;
  int i = t >> 7, f = t & 127;
  float sn = dinv[i] * dinv[i];
  float v = agg[t] + h[t] * sn + bias[f];
  h[t] = fmaxf(v, 0.0f);
}

// ---------------------------------------------------------------------------
// global_mean_pool: batch is SORTED, so accumulate privately over a 64-node
// chunk and flush one atomic per graph-segment (vs one atomic per element).
// Thread layout: f = t & 127 (feature), chunk = t >> 7 (node range).
// ---------------------------------------------------------------------------
__global__ void pool_kernel(const float* __restrict__ h, const int* __restrict__ batch,
                            float* __restrict__ pooled, float* __restrict__ cnts, int N) {
  int t = blockIdx.x * blockDim.x + threadIdx.x;
  int f = t & 127;
  int i0 = (t >> 7) * POOL_CHUNK;
  if (i0 >= N) return;
  int i1 = min(i0 + POOL_CHUNK, N);

  int gcur = batch[i0];
  float acc = 0.0f, cnt = 0.0f;
  for (int i = i0; i < i1; ++i) {
    int g = batch[i];
    if (g != gcur) {                       // sorted -> rare flush
      atomicAdd(&pooled[gcur * FDIM + f], acc);
      if (f == 0) atomicAdd(&cnts[gcur], cnt);
      gcur = g; acc = 0.0f; cnt = 0.0f;
    }
    acc += h[(size_t)i * FDIM + f];
    cnt += 1.0f;
  }
  atomicAdd(&pooled[gcur * FDIM + f], acc);
  if (f == 0) atomicAdd(&cnts[gcur], cnt);
}

// z[g,j] = relu((pooled[g]/cnt[g]) @ fcW1[:,j] + fcb1[j])   (128 x 64)
__global__ void fc1_kernel(const float* __restrict__ pooled, const float* __restrict__ cnts,
                           const float* __restrict__ W, const float* __restrict__ b,
                           float* __restrict__ z) {
  int t = blockIdx.x * blockDim.x + threadIdx.x;
  if (t >= G_NUM * 64) return;
  int g = t >> 6, j = t & 63;
  float inv = 1.0f / fmaxf(cnts[g], 1.0f);
  float s = 0.0f;
  for (int f = 0; f < FDIM; ++f)
    s += pooled[g * FDIM + f] * inv * W[f * 64 + j];
  z[t] = fmaxf(s + b[j], 0.0f);
}

// batch-norm stats per feature (biased var, like torch BN training)
__global__ void bn_stats_kernel(const float* __restrict__ z, float* __restrict__ stats) {
  int j = threadIdx.x;
  if (j >= 64) return;
  float mu = 0.0f, m2 = 0.0f;
  for (int g = 0; g < G_NUM; ++g) {
    float v = z[g * 64 + j];
    mu += v;
    m2 += v * v;
  }
  mu *= (1.0f / G_NUM);
  float var = m2 * (1.0f / G_NUM) - mu * mu;
  stats[j] = mu;
  stats[64 + j] = rsqrtf(var + BN_EPS);
}

// out[g] = (bn(z[g]) * gamma + beta) @ fcW3 + fcb3
__global__ void final_kernel(const float* __restrict__ z, const float* __restrict__ stats,
                             const float* __restrict__ gamma, const float* __restrict__ beta,
                             const float* __restrict__ W3, const float* __restrict__ b3,
                             float* __restrict__ out) {
  int g = threadIdx.x;
  if (g >= G_NUM) return;
  float s = 0.0f;
  for (int j = 0; j < 64; ++j) {
    float v = (z[g * 64 + j] - stats[j]) * stats[64 + j] * gamma[j] + beta[j];
    s += v * W3[j];
  }
  out[g] = s + b3[0];
}

// ---------------------------------------------------------------------------
extern "C" void kernel_launch(void* const* d_in, const int* in_sizes, int n_in,
                              void* d_out, int out_size, void* d_ws, size_t ws_size,
                              hipStream_t stream) {
  const float* x     = (const float*)d_in[0];
  const int*   ei    = (const int*)d_in[1];   // [2,E] int
  const int*   batch = (const int*)d_in[2];
  const float* W1    = (const float*)d_in[3];
  const float* b1    = (const float*)d_in[4];
  const float* W2    = (const float*)d_in[5];
  const float* b2    = (const float*)d_in[6];
  const float* fcW1  = (const float*)d_in[7];
  const float* fcb1  = (const float*)d_in[8];
  const float* gamma = (const float*)d_in[9];
  const float* beta  = (const float*)d_in[10];
  const float* fcW3  = (const float*)d_in[11];
  const float* fcb3  = (const float*)d_in[12];
  float* out = (float*)d_out;

  const int N = in_sizes[0] / FDIM;
  const int E = in_sizes[1] / 2;
  const int* src = ei;
  const int* dst = ei + E;

  // workspace carve-up (keep 16B alignment for float4)
  const int Npad = (N + 3) & ~3;
  float* ws     = (float*)d_ws;
  float* dinv   = ws;                              // N
  float* bufA   = dinv + Npad;                     // N*128 (gemm out / relu out)
  float* bufB   = bufA + (size_t)Npad * FDIM;      // N*128 (aggregation)
  float* pooled = bufB + (size_t)Npad * FDIM;      // 128*128
  float* cnts   = pooled + G_NUM * FDIM;           // 128
  float* zbuf   = cnts + G_NUM;                    // 128*64
  float* stats  = zbuf + G_NUM * 64;               // 128 (mu | rstd)

  dim3 B(256);
  const int nElem   = N * FDIM;
  const int gemmGrd = (N + 127) / 128;
  const int nWaves  = (E + EDGES_PER_WAVE - 1) / EDGES_PER_WAVE;
  const int edgeGrd = (nWaves * 32 + 255) / 256;
  const int elemGrd = (nElem + 255) / 256;
  const int nChunks = (N + POOL_CHUNK - 1) / POOL_CHUNK;
  const int poolGrd = (nChunks * FDIM + 255) / 256;

  // degree -> dinv
  fill_zero<<<256, B, 0, stream>>>(dinv, N);
  degree_kernel<<<(E + 255) / 256, B, 0, stream>>>(dst, dinv, E);
  dinv_kernel<<<(N + 255) / 256, B, 0, stream>>>(dinv, N);

  // layer 1: h = relu(agg(x@W1) + (x@W1)*selfnorm + b1)
  gemm_wmma_f32<<<gemmGrd, B, 0, stream>>>(x, W1, bufA, N);
  fill_zero<<<2048, B, 0, stream>>>(bufB, nElem);
  scatter_kernel<<<edgeGrd, B, 0, stream>>>(src, dst, dinv, bufA, bufB, E);
  combine_kernel<<<elemGrd, B, 0, stream>>>(bufA, bufB, dinv, b1, N);

  // layer 2 (in-place GEMM on bufA)
  gemm_wmma_f32<<<gemmGrd, B, 0, stream>>>(bufA, W2, bufA, N);
  fill_zero<<<2048, B, 0, stream>>>(bufB, nElem);
  scatter_kernel<<<edgeGrd, B, 0, stream>>>(src, dst, dinv, bufA, bufB, E);
  combine_kernel<<<elemGrd, B, 0, stream>>>(bufA, bufB, dinv, b2, N);

  // mean pool + head
  fill_zero<<<64, B, 0, stream>>>(pooled, G_NUM * FDIM + G_NUM);  // pooled + cnts
  pool_kernel<<<poolGrd, B, 0, stream>>>(bufA, batch, pooled, cnts, N);
  fc1_kernel<<<(G_NUM * 64 + 255) / 256, B, 0, stream>>>(pooled, cnts, fcW1, fcb1, zbuf);
  bn_stats_kernel<<<1, 64, 0, stream>>>(zbuf, stats);
  final_kernel<<<1, 128, 0, stream>>>(zbuf, stats, gamma, beta, fcW3, fcb3, out);
}